// GAT_16080357556439
// MI455X (gfx1250) — compile-verified
//
#include <hip/hip_runtime.h>
#include <hip/hip_bf16.h>
#include <math.h>

typedef float v2f __attribute__((ext_vector_type(2)));
typedef float v8f __attribute__((ext_vector_type(8)));

#define DIM 128
#define NEG_SLOPE 0.01f

// ---- 0. init: zero degree + scatter cursors ----
__global__ void gat_init(unsigned* __restrict__ deg, unsigned* __restrict__ cursor, int n) {
    int i = blockIdx.x * blockDim.x + threadIdx.x;
    if (i < n) { deg[i] = 0u; cursor[i] = 0u; }
}

// ---- 1. h = x @ W + b via V_WMMA_F32_16X16X4_F32 ----
// One wave computes a 16-row x 128-col strip of h (8 v8f accumulators).
// W staged in LDS as K-pair-interleaved float2 so each B fragment is one ds_load_b64.
__global__ void __launch_bounds__(256)
gat_gemm_wmma(const float* __restrict__ x, const float* __restrict__ W,
              const float* __restrict__ bias, float* __restrict__ h, int n_rows) {
    __shared__ v2f Wp[64 * DIM];  // Wp[p*128+col] = {W[2p][col], W[2p+1][col]}, 64 KB
    for (int i = threadIdx.x; i < 64 * DIM; i += blockDim.x) {
        const int p = i >> 7, col = i & 127;
        v2f w;
        w.x = W[(2 * p) * DIM + col];
        w.y = W[(2 * p + 1) * DIM + col];
        Wp[i] = w;
    }
    __syncthreads();

    const int wave  = threadIdx.x >> 5;
    const int lane  = threadIdx.x & 31;
    const int strip = blockIdx.x * (blockDim.x >> 5) + wave;
    const int row0  = strip * 16;
    if (row0 >= n_rows) return;            // wave-uniform exit (EXEC all-1 inside)

    const int lo = lane & 15;              // M (A) / N (B, C)
    const int hi = lane >> 4;              // K-pair select / C row half

    v8f c[8];
#pragma unroll
    for (int t = 0; t < 8; ++t) c[t] = (v8f)(0.0f);

    int arow = row0 + lo; if (arow >= n_rows) arow = n_rows - 1;
    const float* xrow = x + (size_t)arow * DIM;

    for (int k0 = 0; k0 < DIM; k0 += 4) {
        v2f a = *(const v2f*)(xrow + k0 + 2 * hi);    // A: 16x4 fragment (b64 load)
        const int pa = (k0 >> 1) + hi;                // K-pair index
#pragma unroll
        for (int t = 0; t < 8; ++t) {
            v2f bf = Wp[pa * DIM + t * 16 + lo];      // B: 4x16 fragment (one ds_load_b64)
            c[t] = __builtin_amdgcn_wmma_f32_16x16x4_f32(
                false, a, false, bf, (short)0, c[t], false, false);
        }
    }

    // Epilogue: VGPR j holds row (j + 8*hi), lane lo = col within tile.
#pragma unroll
    for (int t = 0; t < 8; ++t) {
        const int col = t * 16 + lo;
        const float bc = bias[col];
#pragma unroll
        for (int j = 0; j < 8; ++j) {
            const int row = row0 + j + 8 * hi;
            if (row < n_rows) h[(size_t)row * DIM + col] = c[t][j] + bc;
        }
    }
}

// ---- 2. u[i] = dot(h[i], a[0:128]); v[i] = dot(h[i], a[128:256]) ----
__global__ void gat_uv(const float* __restrict__ h, const float* __restrict__ a,
                       float* __restrict__ u, float* __restrict__ v, int n) {
    const int node = (int)((blockIdx.x * (size_t)blockDim.x + threadIdx.x) >> 5);
    const int lane = threadIdx.x & 31;
    if (node >= n) return;
    float4 hv = ((const float4*)(h + (size_t)node * DIM))[lane];
    float4 as = ((const float4*)a)[lane];
    float4 ad = ((const float4*)(a + DIM))[lane];
    float su = hv.x * as.x + hv.y * as.y + hv.z * as.z + hv.w * as.w;
    float sv = hv.x * ad.x + hv.y * ad.y + hv.z * ad.z + hv.w * ad.w;
#pragma unroll
    for (int off = 16; off > 0; off >>= 1) {
        su += __shfl_xor(su, off, 32);
        sv += __shfl_xor(sv, off, 32);
    }
    if (lane == 0) { u[node] = su; v[node] = sv; }
}

// ---- 3. degree histogram over src (edges + self-loops) ----
__global__ void gat_hist(const int* __restrict__ ei, unsigned* __restrict__ deg,
                         int ne, int t) {
    int e = blockIdx.x * blockDim.x + threadIdx.x;
    if (e >= t) return;
    int s = (e < ne) ? ei[e] : (e - ne);
    atomicAdd(deg + s, 1u);
}

// ---- 4. exclusive scan deg -> row_ptr (single block, chunked Hillis-Steele) ----
__global__ void __launch_bounds__(256)
gat_scan(const unsigned* __restrict__ deg, unsigned* __restrict__ row_ptr, int n) {
    __shared__ unsigned buf[256];
    __shared__ unsigned carry_s;
    const int tid = threadIdx.x;
    if (tid == 0) carry_s = 0u;
    __syncthreads();
    for (int base = 0; base < n; base += 256) {
        const int i = base + tid;
        const unsigned val = (i < n) ? deg[i] : 0u;
        buf[tid] = val;
        __syncthreads();
        for (int off = 1; off < 256; off <<= 1) {
            unsigned t2 = (tid >= off) ? buf[tid - off] : 0u;
            __syncthreads();
            buf[tid] += t2;
            __syncthreads();
        }
        const unsigned carry = carry_s;
        if (i < n) row_ptr[i] = carry + buf[tid] - val;  // exclusive
        __syncthreads();
        if (tid == 255) carry_s = carry + buf[255];
        __syncthreads();
    }
    if (tid == 0) row_ptr[n] = carry_s;
}

// ---- 5. scatter dst indices into CSR order ----
__global__ void gat_scatter(const int* __restrict__ ei, const unsigned* __restrict__ row_ptr,
                            unsigned* __restrict__ cursor, int* __restrict__ dst_sorted,
                            int ne, int t) {
    int e = blockIdx.x * blockDim.x + threadIdx.x;
    if (e >= t) return;
    int s, d;
    if (e < ne) { s = ei[e]; d = ei[ne + e]; } else { s = d = e - ne; }
    unsigned pos = row_ptr[s] + atomicAdd(cursor + s, 1u);
    dst_sorted[pos] = d;
}

// ---- 6. fused per-node softmax + weighted aggregation + ELU ----
// One wave per node; zero float atomics; h gather stays in the 192MB L2.
__global__ void gat_node_aggr(const float* __restrict__ h, const float* __restrict__ u,
                              const float* __restrict__ v, const float* __restrict__ a_bias,
                              const unsigned* __restrict__ row_ptr,
                              const int* __restrict__ dst_sorted,
                              float* __restrict__ out, int n) {
    const int node = (int)((blockIdx.x * (size_t)blockDim.x + threadIdx.x) >> 5);
    const int lane = threadIdx.x & 31;
    if (node >= n) return;
    const int start = (int)row_ptr[node];
    const int end   = (int)row_ptr[node + 1];
    const float base = u[node] + a_bias[0];

    // phase 1: segment max (lane-strided, then wave reduce)
    float m = -INFINITY;
    for (int j = start + lane; j < end; j += 32) {
        float l = base + v[dst_sorted[j]];
        l = (l > 0.0f) ? l : NEG_SLOPE * l;
        m = fmaxf(m, l);
    }
#pragma unroll
    for (int off = 16; off > 0; off >>= 1) m = fmaxf(m, __shfl_xor(m, off, 32));

    // phase 2: softmax denominator
    float s = 0.0f;
    for (int j = start + lane; j < end; j += 32) {
        float l = base + v[dst_sorted[j]];
        l = (l > 0.0f) ? l : NEG_SLOPE * l;
        s += expf(l - m);
    }
#pragma unroll
    for (int off = 16; off > 0; off >>= 1) s += __shfl_xor(s, off, 32);
    const float inv_denom = 1.0f / s;

    // phase 3: whole wave per edge; lane owns 4 output columns
    float4 acc = {0.0f, 0.0f, 0.0f, 0.0f};
    for (int j = start; j < end; ++j) {
        const int d = dst_sorted[j];
        float l = base + v[d];                   // broadcast scalar load
        l = (l > 0.0f) ? l : NEG_SLOPE * l;
        const float alpha = expf(l - m) * inv_denom;
        float4 hv = ((const float4*)(h + (size_t)d * DIM))[lane];
        acc.x += alpha * hv.x;
        acc.y += alpha * hv.y;
        acc.z += alpha * hv.z;
        acc.w += alpha * hv.w;
    }

    // fused ELU + single coalesced store per row
    acc.x = (acc.x > 0.0f) ? acc.x : expf(acc.x) - 1.0f;
    acc.y = (acc.y > 0.0f) ? acc.y : expf(acc.y) - 1.0f;
    acc.z = (acc.z > 0.0f) ? acc.z : expf(acc.z) - 1.0f;
    acc.w = (acc.w > 0.0f) ? acc.w : expf(acc.w) - 1.0f;
    ((float4*)(out + (size_t)node * DIM))[lane] = acc;
}

extern "C" void kernel_launch(void* const* d_in, const int* in_sizes, int n_in,
                              void* d_out, int out_size, void* d_ws, size_t ws_size,
                              hipStream_t stream) {
    const float* x      = (const float*)d_in[0];
    const int*   ei     = (const int*)  d_in[1];
    const float* W      = (const float*)d_in[2];
    const float* b      = (const float*)d_in[3];
    const float* a      = (const float*)d_in[4];
    const float* a_bias = (const float*)d_in[5];
    float* out = (float*)d_out;

    const int n  = in_sizes[0] / DIM;   // 50000 nodes
    const int ne = in_sizes[1] / 2;     // 1.6M edges
    const int t  = ne + n;              // + self-loops

    char* ws = (char*)d_ws;
    float*    h          = (float*)ws;    ws += (size_t)n * DIM * sizeof(float);
    float*    u          = (float*)ws;    ws += (size_t)n * sizeof(float);
    float*    v          = (float*)ws;    ws += (size_t)n * sizeof(float);
    unsigned* deg        = (unsigned*)ws; ws += (size_t)n * sizeof(unsigned);
    unsigned* cursor     = (unsigned*)ws; ws += (size_t)n * sizeof(unsigned);
    unsigned* row_ptr    = (unsigned*)ws; ws += (size_t)(n + 1) * sizeof(unsigned);
    int*      dst_sorted = (int*)ws;

    // 0. zero counters
    gat_init<<<(n + 255) / 256, 256, 0, stream>>>(deg, cursor, n);

    // 1. GEMM h = xW + b (8 waves/block, one 16-row strip per wave)
    {
        int strips = (n + 15) / 16;
        gat_gemm_wmma<<<(strips + 7) / 8, 256, 0, stream>>>(x, W, b, h, n);
    }

    // 2. u, v per node (wave per node)
    gat_uv<<<(n + 7) / 8, 256, 0, stream>>>(h, a, u, v, n);

    // 3-5. CSR build: histogram -> exclusive scan -> scatter
    gat_hist<<<(t + 255) / 256, 256, 0, stream>>>(ei, deg, ne, t);
    gat_scan<<<1, 256, 0, stream>>>(deg, row_ptr, n);
    gat_scatter<<<(t + 255) / 256, 256, 0, stream>>>(ei, row_ptr, cursor, dst_sorted, ne, t);

    // 6. fused softmax + aggregation + ELU (wave per node, no float atomics)
    gat_node_aggr<<<(n + 7) / 8, 256, 0, stream>>>(h, u, v, a_bias, row_ptr, dst_sorted, out, n);
}